// MixedLinear_86139864088895
// MI455X (gfx1250) — compile-verified
//
#include <hip/hip_runtime.h>

#define MAX_IN  1024
#define MAX_OUT 4096
#define MROWS   8192   // B * S = 4 * 2048

typedef __attribute__((ext_vector_type(2))) float v2f;
typedef __attribute__((ext_vector_type(4))) float v4f;
typedef __attribute__((ext_vector_type(8))) float v8f;
typedef __attribute__((ext_vector_type(4))) int   v4i;

__device__ __forceinline__ float fq(float v, float inv_s, float s, float qn, float qp) {
    // clip(v/s, qn, qp), round-half-even, rescale  (matches jnp.round RNE)
    float t = v * inv_s;
    t = fminf(fmaxf(t, qn), qp);
    return rintf(t) * s;
}

// 16-byte async global->LDS copy (GLOBAL_LOAD_ASYNC_TO_LDS_B128, ASYNCcnt).
// Builtin parameter types (from compiler diagnostic): int4* in global AS,
// int4* in LDS AS, imm offset, imm cpol.
__device__ __forceinline__ void async_copy16(const float* g, float* l) {
    __builtin_amdgcn_global_load_async_to_lds_b128(
        (__attribute__((address_space(1))) v4i*)g,
        (__attribute__((address_space(3))) v4i*)l,
        /*imm offset=*/0, /*cpol=*/0);
}

// ---------------------------------------------------------------------------
// Coefficient kernel: collapse the 16 arch weights into 12 closed-form scalars
// c[0..1]  = Ca4, Ca8                     (x_mix coefficients)
// c[2..5]  = Cw4 per quadrant q=0..3      (q = 2*(i>=768) + (o>=3072))
// c[6..9]  = Cw8 per quadrant
// c[10..11]= Cb (o<3072), Cb (o>=3072)    (bias coefficients)
// ---------------------------------------------------------------------------
__global__ void coef_kernel(const float* __restrict__ aw, float* __restrict__ c) {
    if (threadIdx.x != 0 || blockIdx.x != 0) return;
    float a[16];
    #pragma unroll
    for (int k = 0; k < 16; ++k) a[k] = aw[k];

    float ca0 = 0.f, ca1 = 0.f, cb_all = 0.f, cb_it1 = 0.f;
    float S[2][2][2] = {};  // [n][hi][iti]
    #pragma unroll
    for (int hi = 0; hi < 2; ++hi)
        #pragma unroll
        for (int iti = 0; iti < 2; ++iti)
            #pragma unroll
            for (int m = 0; m < 2; ++m)
                #pragma unroll
                for (int n = 0; n < 2; ++n) {
                    int k = hi * 8 + iti * 4 + m * 2 + n;   // reference loop order
                    float w = a[k];
                    if (m == 0) ca0 += w; else ca1 += w;
                    S[n][hi][iti] += w;
                    cb_all += w;
                    if (iti == 1) cb_it1 += w;
                }
    c[0] = ca0;
    c[1] = ca1;
    #pragma unroll
    for (int n = 0; n < 2; ++n) {
        float q0 = S[n][0][0] + S[n][0][1] + S[n][1][0] + S[n][1][1]; // i<768, o<3072
        float q1 = S[n][0][1] + S[n][1][1];                           // i<768, o>=3072
        float q2 = S[n][1][0] + S[n][1][1];                           // i>=768, o<3072
        float q3 = S[n][1][1];                                        // i>=768, o>=3072
        c[2 + n * 4 + 0] = q0;
        c[2 + n * 4 + 1] = q1;
        c[2 + n * 4 + 2] = q2;
        c[2 + n * 4 + 3] = q3;
    }
    c[10] = cb_all;
    c[11] = cb_it1;
}

// ---------------------------------------------------------------------------
// w_mix[o,i] = Cw4(q)*fq4(w) + Cw8(q)*fq8(w)    (float4 per thread)
// Raw weight is read once -> NT load; w_mix is reused by GEMM -> RT store.
// ---------------------------------------------------------------------------
__global__ void wmix_kernel(const float* __restrict__ w,
                            const float* __restrict__ w_sc,
                            const float* __restrict__ c,
                            float* __restrict__ wmix) {
    const int gid = blockIdx.x * blockDim.x + threadIdx.x;
    const int idx = gid * 4;
    const int o = idx >> 10;          // /1024
    const int i = idx & 1023;         // %1024 (768 % 4 == 0: no straddle)
    const float s4 = w_sc[0], s8 = w_sc[1];
    const float i4 = 1.0f / s4, i8 = 1.0f / s8;
    const int q = ((i >= 768) ? 2 : 0) + ((o >= 3072) ? 1 : 0);
    const float c4 = c[2 + q];
    const float c8 = c[6 + q];
    v4f wv = __builtin_nontemporal_load((const v4f*)(w + idx));
    v4f r;
    #pragma unroll
    for (int t = 0; t < 4; ++t) {
        float f4 = fq(wv[t], i4, s4, -8.0f, 7.0f);
        float f8 = fq(wv[t], i8, s8, -128.0f, 127.0f);
        r[t] = c4 * f4 + c8 * f8;
    }
    *(v4f*)(wmix + idx) = r;
}

// ---------------------------------------------------------------------------
// x_mix = Ca4*fq4(x) + Ca8*fq8(x)               (float4 per thread)
// ---------------------------------------------------------------------------
__global__ void xmix_kernel(const float* __restrict__ x,
                            const float* __restrict__ a_sc,
                            const float* __restrict__ c,
                            float* __restrict__ xmix) {
    const int gid = blockIdx.x * blockDim.x + threadIdx.x;
    const int idx = gid * 4;
    const float s4 = a_sc[0], s8 = a_sc[1];
    const float i4 = 1.0f / s4, i8 = 1.0f / s8;
    const float ca4 = c[0], ca8 = c[1];
    v4f xv = __builtin_nontemporal_load((const v4f*)(x + idx));
    v4f r;
    #pragma unroll
    for (int t = 0; t < 4; ++t) {
        float f4 = fq(xv[t], i4, s4, -8.0f, 7.0f);
        float f8 = fq(xv[t], i8, s8, -128.0f, 127.0f);
        r[t] = ca4 * f4 + ca8 * f8;
    }
    *(v4f*)(xmix + idx) = r;
}

// ---------------------------------------------------------------------------
// GEMM: out[m,n] = sum_k xmix[m,k]*wmix[n,k] + bias[n]*Cb(n)
// 256 threads = 8 wave32, block tile 128x128, BK=32, V_WMMA_F32_16X16X4_F32.
// Waves tiled 4(M) x 2(N); each wave: 32x64 = 2x4 tiles of 16x16 (8 x v8f acc).
// Double-buffered LDS filled by GLOBAL_LOAD_ASYNC_TO_LDS_B128 (ASYNCcnt).
// ---------------------------------------------------------------------------
#define BM   128
#define BN   128
#define BK   32
#define LDSP 36   // padded row stride (floats): 8B-aligned, spreads banks

__launch_bounds__(256, 2)
__global__ void gemm_kernel(const float* __restrict__ xm,
                            const float* __restrict__ wm,
                            const float* __restrict__ bias,
                            const float* __restrict__ c,
                            float* __restrict__ out) {
    __shared__ float As[2][BM * LDSP];
    __shared__ float Bs[2][BN * LDSP];

    const int tid  = threadIdx.x;
    const int wave = tid >> 5;
    const int lane = tid & 31;
    const int half = lane >> 4;      // 0: lanes 0-15, 1: lanes 16-31
    const int lm   = lane & 15;
    const int wmi  = wave >> 1;      // 0..3 -> M sub-tile
    const int wni  = wave & 1;       // 0..1 -> N sub-tile
    const int bm   = blockIdx.y * BM;
    const int bn   = blockIdx.x * BN;

    const int lr = tid >> 3;         // 0..31 : row within 32-row group
    const int lc = (tid & 7) * 4;    // 0,4,...,28 : k offset (float4)

    // Issue one 128x32 tile of A and B into buffer b (8 async b128 per thread).
    auto issue = [&](int b, int kt2) {
        #pragma unroll
        for (int p = 0; p < 4; ++p) {
            const int r = p * 32 + lr;
            async_copy16(xm + (size_t)(bm + r) * MAX_IN + kt2 + lc, &As[b][r * LDSP + lc]);
            async_copy16(wm + (size_t)(bn + r) * MAX_IN + kt2 + lc, &Bs[b][r * LDSP + lc]);
        }
    };

    v8f acc[2][4];
    #pragma unroll
    for (int mt = 0; mt < 2; ++mt)
        #pragma unroll
        for (int nt = 0; nt < 4; ++nt)
            acc[mt][nt] = (v8f){0.f, 0.f, 0.f, 0.f, 0.f, 0.f, 0.f, 0.f};

    int buf = 0;
    issue(0, 0);   // prologue fill

    for (int kt = 0; kt < MAX_IN; kt += BK) {
        const bool more = (kt + BK < MAX_IN);
        if (more) {
            issue(buf ^ 1, kt + BK);                 // prefetch next tile
            __builtin_amdgcn_s_wait_asynccnt(8);     // current tile's 8 done
        } else {
            __builtin_amdgcn_s_wait_asynccnt(0);
        }
        __syncthreads();                             // tile visible to all waves

        const float* Ab = &As[buf][0];
        const float* Bb = &Bs[buf][0];
        #pragma unroll
        for (int kk = 0; kk < BK; kk += 4) {
            // ISA 16x4 f32 A layout: lane row = lane%16, K = kk + 2*(lane/16) + {0,1}
            v2f afrag[2], bfrag[4];
            #pragma unroll
            for (int mt = 0; mt < 2; ++mt)
                afrag[mt] = *(const v2f*)(Ab + (wmi * 32 + mt * 16 + lm) * LDSP + kk + 2 * half);
            #pragma unroll
            for (int nt = 0; nt < 4; ++nt)
                bfrag[nt] = *(const v2f*)(Bb + (wni * 64 + nt * 16 + lm) * LDSP + kk + 2 * half);
            #pragma unroll
            for (int mt = 0; mt < 2; ++mt)
                #pragma unroll
                for (int nt = 0; nt < 4; ++nt)
                    acc[mt][nt] = __builtin_amdgcn_wmma_f32_16x16x4_f32(
                        /*neg_a=*/false, afrag[mt],
                        /*neg_b=*/false, bfrag[nt],
                        /*c_mod=*/(short)0, acc[mt][nt],
                        /*reuse_a=*/false, /*reuse_b=*/false);
        }
        __syncthreads();                             // done reading buf: safe to refill
        buf ^= 1;
    }

    // Epilogue: C/D layout = VGPR r, lane l -> M = r + 8*(l>=16), N = l%16
    // Output is write-once (134 MB): NT stores keep x_mix/w_mix resident in L2.
    const float cb0 = c[10], cb1 = c[11];
    #pragma unroll
    for (int mt = 0; mt < 2; ++mt) {
        #pragma unroll
        for (int nt = 0; nt < 4; ++nt) {
            const int gn = bn + wni * 64 + nt * 16 + lm;
            const float bv = bias[gn] * (gn < 3072 ? cb0 : cb1);
            const int m0 = bm + wmi * 32 + mt * 16 + 8 * half;
            #pragma unroll
            for (int r = 0; r < 8; ++r)
                __builtin_nontemporal_store(acc[mt][nt][r] + bv,
                                            out + (size_t)(m0 + r) * MAX_OUT + gn);
        }
    }
}

extern "C" void kernel_launch(void* const* d_in, const int* in_sizes, int n_in,
                              void* d_out, int out_size, void* d_ws, size_t ws_size,
                              hipStream_t stream) {
    (void)in_sizes; (void)n_in; (void)out_size; (void)ws_size;
    const float* x    = (const float*)d_in[0];   // [4,2048,1024]
    const float* aw   = (const float*)d_in[1];   // [16]
    const float* w    = (const float*)d_in[2];   // [4096,1024]
    const float* bias = (const float*)d_in[3];   // [4096]
    const float* a_sc = (const float*)d_in[4];   // [2]
    const float* w_sc = (const float*)d_in[5];   // [2]
    float* out = (float*)d_out;                  // [4,2048,4096]

    float* wsf  = (float*)d_ws;
    float* coef = wsf;                              // 12 floats (64-slot pad)
    float* wmix = wsf + 64;                         // 4096*1024 floats
    float* xmix = wmix + (size_t)MAX_OUT * MAX_IN;  // 8192*1024 floats

    coef_kernel<<<dim3(1), dim3(32), 0, stream>>>(aw, coef);
    wmix_kernel<<<dim3((MAX_OUT * MAX_IN / 4) / 256), dim3(256), 0, stream>>>(w, w_sc, coef, wmix);
    xmix_kernel<<<dim3((MROWS * MAX_IN / 4) / 256), dim3(256), 0, stream>>>(x, a_sc, coef, xmix);
    gemm_kernel<<<dim3(MAX_OUT / BN, MROWS / BM), dim3(256), 0, stream>>>(xmix, wmix, bias, coef, out);
}